// pseudo_loss_64433099375147
// MI455X (gfx1250) — compile-verified
//
#include <hip/hip_runtime.h>

// Problem constants (match reference).
#define CUR_THR 0.7f
constexpr int B = 8, N = 2048, M = 2048;
constexpr int TPB      = 256;            // 8 wave32s per block
constexpr int GT_CHUNK = 256;            // gt boxes staged per block (4 KB LDS)
constexpr int NCHUNK   = M / GT_CHUNK;   // 8 partial-max slices per pred

typedef float v2f __attribute__((ext_vector_type(2)));
typedef float v8f __attribute__((ext_vector_type(8)));

// One-instruction lane XOR exchange: ds_swizzle group-of-32 mode,
// offset = {1'b0, and_mask=0x1f, or_mask=0, xor_mask} = 0x7C00 | xor_mask.
#define LANE_XOR_ADD(s, mask)                                                  \
  (s) += __int_as_float(                                                       \
      __builtin_amdgcn_ds_swizzle(__float_as_int(s), 0x7C00 | (mask)))

// Wave32 sum-reduction through the matrix pipe:
// D = ones(16x4) x B with B vgpr0 = lane value, vgpr1 = 0.
// 16x16x4 f32 B layout: lanes 0-15 carry K=0/1, lanes 16-31 carry K=2/3,
// N = lane&15  =>  D[.][n] = v_n + v_{n+16} replicated to every row, so lane l
// holds colsum_{l&15}; 4 ds_swizzle XOR steps finish the 16-way sum.
__device__ __forceinline__ float wave_reduce_sum_wmma(float v) {
  v2f a; a.x = 1.0f; a.y = 1.0f;     // A = ones
  v2f b; b.x = v;    b.y = 0.0f;     // only K=0 / K=2 carry data
  v8f c = {};
  c = __builtin_amdgcn_wmma_f32_16x16x4_f32(false, a, false, b, (short)0, c,
                                            false, false);
  float s = c[0];
  LANE_XOR_ADD(s, 1);
  LANE_XOR_ADD(s, 2);
  LANE_XOR_ADD(s, 4);
  LANE_XOR_ADD(s, 8);
  return s;  // valid in all lanes
}

// ---------------------------------------------------------------------------
// Kernel 1: per-image weight = mean(1 - (s - thr)/(1 - thr)); also zero d_out.
// ---------------------------------------------------------------------------
__global__ void __launch_bounds__(TPB)
weight_kernel(const float* __restrict__ scores, float* __restrict__ ws_w,
              float* __restrict__ out) {
  const int b = blockIdx.x, tid = threadIdx.x;
  float s = 0.0f;
#pragma unroll
  for (int j = 0; j < M / TPB; ++j) {  // uniform trip count: EXEC stays all-1s
    float x = scores[(size_t)b * M + j * TPB + tid];
    s += 1.0f - (x - CUR_THR) * (1.0f / (1.0f - CUR_THR));
  }
  s = wave_reduce_sum_wmma(s);
  __shared__ float red[TPB / 32];
  const int wave = tid >> 5, lane = tid & 31;
  if (lane == 0) red[wave] = s;
  __syncthreads();
  if (tid == 0) {
    float t = 0.0f;
#pragma unroll
    for (int i = 0; i < TPB / 32; ++i) t += red[i];
    ws_w[b] = t * (1.0f / (float)M);
    if (b == 0) out[0] = 0.0f;  // runs before finalize_kernel (stream order)
  }
}

// ---------------------------------------------------------------------------
// Kernel 2: partial row-max of IoU. Grid = B * (N/TPB) * NCHUNK = 512 blocks.
// One thread = one pred box vs one 256-box gt slice staged in LDS via the
// CDNA5 async global->LDS path. The "+1" of the VOC IoU is folded into the
// staged x2/y2 so the inner loop does 2 fewer VALU ops per pair.
// ---------------------------------------------------------------------------
__global__ void __launch_bounds__(TPB)
iou_partial_kernel(const float* __restrict__ pred, const float* __restrict__ gt,
                   float* __restrict__ ws_max) {
  __shared__ __align__(16) float sg[GT_CHUNK * 4];
  __shared__ float sarea[GT_CHUNK];

  const int tid   = threadIdx.x;
  const int blk   = blockIdx.x;
  const int chunk = blk & (NCHUNK - 1);
  const int ptile = (blk / NCHUNK) & (N / TPB - 1);
  const int b     = blk / (NCHUNK * (N / TPB));
  const int p     = ptile * TPB + tid;

  // Async-copy this block's 256 gt boxes (4 KB) into LDS: 256 lanes x b128.
  {
    const float* gsrc =
        gt + ((size_t)b * M + (size_t)chunk * GT_CHUNK) * 4 + (size_t)tid * 4;
    unsigned lds_off = (unsigned)(size_t)(&sg[tid * 4]);  // low 32b = LDS offset
    unsigned long long gaddr = (unsigned long long)(size_t)gsrc;
    asm volatile("global_load_async_to_lds_b128 %0, %1, off"
                 :: "v"(lds_off), "v"(gaddr)
                 : "memory");
    asm volatile("s_wait_asynccnt 0x0" ::: "memory");
  }
  __syncthreads();

  // Precompute gt areas and fold the "+1" into x2/y2 in place.
  {
    float x1 = sg[tid * 4 + 0], y1 = sg[tid * 4 + 1];
    float x2 = sg[tid * 4 + 2], y2 = sg[tid * 4 + 3];
    sarea[tid] = (x2 - x1 + 1.0f) * (y2 - y1 + 1.0f);
    sg[tid * 4 + 2] = x2 + 1.0f;
    sg[tid * 4 + 3] = y2 + 1.0f;
  }
  __syncthreads();

  const float4 pbx = ((const float4*)pred)[(size_t)b * N + p];
  const float px1 = pbx.x, py1 = pbx.y;
  const float px2p = pbx.z + 1.0f, py2p = pbx.w + 1.0f;  // folded "+1"
  const float ap   = (px2p - px1) * (py2p - py1);

  // Two independent max chains (even/odd j) to keep v_max off the critical path.
  float m0 = 0.0f, m1 = 0.0f;
#pragma unroll 4
  for (int j = 0; j < GT_CHUNK; j += 2) {
#pragma unroll
    for (int u = 0; u < 2; ++u) {
      const int jj = j + u;
      // Broadcast LDS reads (all lanes same address -> conflict-free).
      float gx1  = sg[jj * 4 + 0], gy1  = sg[jj * 4 + 1];
      float gx2p = sg[jj * 4 + 2], gy2p = sg[jj * 4 + 3];
      float w = fminf(px2p, gx2p) - fmaxf(px1, gx1);
      float h = fminf(py2p, gy2p) - fmaxf(py1, gy1);
      w = fmaxf(w, 0.0f);
      h = fmaxf(h, 0.0f);
      float inter = w * h;
      float denom = (ap + sarea[jj]) - inter;
      float iou   = inter * __builtin_amdgcn_rcpf(denom);  // trans pipe co-exec
      if (u == 0) m0 = fmaxf(m0, iou); else m1 = fmaxf(m1, iou);
    }
  }
  ws_max[((size_t)b * N + p) * NCHUNK + chunk] = fmaxf(m0, m1);
}

// ---------------------------------------------------------------------------
// Kernel 3: fold the NCHUNK partial maxes, weight, and globally reduce.
// Grid = B*N/TPB = 64 blocks; each block lies inside one image (2048 % 256==0).
// ---------------------------------------------------------------------------
__global__ void __launch_bounds__(TPB)
finalize_kernel(const float* __restrict__ ws_max, const float* __restrict__ ws_w,
                float* __restrict__ out) {
  const int gid = blockIdx.x * TPB + threadIdx.x;  // pred index across batch
  const int b   = gid / N;
  // 8 partial maxes = 32 bytes, 32B-aligned: two b128 loads.
  const float4* w4 = (const float4*)(ws_max + (size_t)gid * NCHUNK);
  float4 q0 = w4[0], q1 = w4[1];
  float mx = fmaxf(fmaxf(fmaxf(q0.x, q0.y), fmaxf(q0.z, q0.w)),
                   fmaxf(fmaxf(q1.x, q1.y), fmaxf(q1.z, q1.w)));
  float contrib = ws_w[b] * (1.0f - mx);

  float s = wave_reduce_sum_wmma(contrib);
  __shared__ float red[TPB / 32];
  const int wave = threadIdx.x >> 5, lane = threadIdx.x & 31;
  if (lane == 0) red[wave] = s;
  __syncthreads();
  if (threadIdx.x == 0) {
    float t = 0.0f;
#pragma unroll
    for (int i = 0; i < TPB / 32; ++i) t += red[i];
    atomicAdd(out, t);  // global_atomic_add_f32
  }
}

// ---------------------------------------------------------------------------
extern "C" void kernel_launch(void* const* d_in, const int* in_sizes, int n_in,
                              void* d_out, int out_size, void* d_ws,
                              size_t ws_size, hipStream_t stream) {
  (void)in_sizes; (void)n_in; (void)out_size; (void)ws_size;
  const float* pred   = (const float*)d_in[0];  // [B,N,4]
  const float* gt     = (const float*)d_in[1];  // [B,M,4]
  const float* scores = (const float*)d_in[2];  // [B,M]
  float* out    = (float*)d_out;                // [1]
  float* ws_max = (float*)d_ws;                        // B*N*NCHUNK floats (512 KB)
  float* ws_w   = ws_max + (size_t)B * N * NCHUNK;     // B floats

  weight_kernel<<<B, TPB, 0, stream>>>(scores, ws_w, out);
  iou_partial_kernel<<<B * (N / TPB) * NCHUNK, TPB, 0, stream>>>(pred, gt, ws_max);
  finalize_kernel<<<(B * N) / TPB, TPB, 0, stream>>>(ws_max, ws_w, out);
}